// GraphConvNetwork_12532714569913
// MI455X (gfx1250) — compile-verified
//
#include <hip/hip_runtime.h>
#include <math.h>

// ---------------------------------------------------------------------------
// 3-layer GCN for MI455X (gfx1250, wave32).
//  - GEMMs: V_WMMA_F32_16X16X4_F32, fp32 accumulate (matches fp32 reference).
//    Weights are transposed+padded once so every fragment load is an
//    unconditional global_load_b64 (no exec-mask branches).
//  - Edge aggregation: wave-per-edge, f32 global atomics. h (12.8MB) and agg
//    (12.8MB) both sit in the 192MB L2, so the scatter runs at L2 atomic
//    rate; HBM only sees the edge list + one pass over node arrays.
// ---------------------------------------------------------------------------

#define N_NODES   50000
#define D_FEAT    64
#define N_CLASSES 40
#define TILES_M   (N_NODES / 16)   // 3125, exact

typedef float v2f __attribute__((ext_vector_type(2)));
typedef float v8f __attribute__((ext_vector_type(8)));

// -------------------------------- utility ----------------------------------

__global__ void zero_f32(float* __restrict__ p, int n) {
  int i = blockIdx.x * blockDim.x + threadIdx.x;
  if (i < n) p[i] = 0.0f;
}

// deg[dst] += 1 over real edges; self-loop added analytically in dinv_kernel
__global__ void deg_kernel(const int* __restrict__ dst, float* __restrict__ deg, int E) {
  int i = blockIdx.x * blockDim.x + threadIdx.x;
  if (i < E) atomicAdd(&deg[dst[i]], 1.0f);
}

// in-place deg -> dinv = rsqrt(deg + 1)   (the +1 is the self-loop)
__global__ void dinv_kernel(float* __restrict__ d, int n) {
  int i = blockIdx.x * blockDim.x + threadIdx.x;
  if (i < n) d[i] = rsqrtf(d[i] + 1.0f);
}

// Wt[n*64 + k] = (n < F) ? W[k*F + n] : 0   for n in [0, Fpad), k in [0, 64)
__global__ void transpose_pad_w(const float* __restrict__ W, float* __restrict__ Wt,
                                int F, int Fpad) {
  int i = blockIdx.x * blockDim.x + threadIdx.x;
  if (i >= Fpad * D_FEAT) return;
  int n = i >> 6;          // i / 64
  int k = i & 63;          // i % 64
  Wt[i] = (n < F) ? W[k * F + n] : 0.0f;
}

// ------------------------------- WMMA GEMM ---------------------------------
// H[N_NODES x Fpad] = A[N_NODES x 64] * Wt^T, H row stride 64 always.
// One wave = one 16x16 C tile; 16 chained V_WMMA_F32_16X16X4_F32.
// Branch-free body: A and B fragment loads are both contiguous b64 loads.
__global__ void gemm_wmma(const float* __restrict__ A, const float* __restrict__ Wt,
                          float* __restrict__ H, int tilesN) {
  const int lane  = threadIdx.x & 31;
  const int gw    = (blockIdx.x * blockDim.x + threadIdx.x) >> 5;
  const int total = TILES_M * tilesN;
  if (gw >= total) return;                 // whole-wave exit; EXEC stays full

  const int mtile = gw % TILES_M;
  const int ntile = gw / TILES_M;
  const int m0 = mtile * 16;
  const int n0 = ntile * 16;

  const int g   = lane >> 4;               // half-wave group (0 or 1)
  const int row = m0 + (lane & 15);
  const int col = n0 + (lane & 15);        // col <= 47 < 64 even for layer 3

  const float* __restrict__ arow = A  + (size_t)row * D_FEAT;
  const float* __restrict__ bcol = Wt + (size_t)col * D_FEAT;

  v8f c = {};
  #pragma unroll
  for (int k0 = 0; k0 < D_FEAT; k0 += 4) {
    const int ka = k0 + 2 * g;             // even -> 8B-aligned b64 loads
    v2f a = *reinterpret_cast<const v2f*>(arow + ka);
    v2f b = *reinterpret_cast<const v2f*>(bcol + ka);
    c = __builtin_amdgcn_wmma_f32_16x16x4_f32(
        /*neg_a=*/false, a, /*neg_b=*/false, b,
        /*c_mod=*/(short)0, c, /*reuse_a=*/false, /*reuse_b=*/false);
  }

  // C/D layout: VGPR j -> rows m0+j (lanes 0-15) and m0+j+8 (lanes 16-31).
  // Unconditional store: H row stride is 64 and col < 64 always.
  #pragma unroll
  for (int j = 0; j < 8; ++j) {
    H[(size_t)(m0 + j + 8 * g) * D_FEAT + col] = c[j];
  }
}

// ------------------------------ edge scatter -------------------------------
// One wave per edge: lane pairs cover the feature vector (2 floats/lane for
// F=64; lanes 0..19 for F=40). src/dst are wave-uniform -> scalar loads; the
// gather of h[src] is two consecutive 128B lines; adds are f32 L2 atomics.
template <int F>
__global__ void scatter_kernel(const float* __restrict__ h,
                               const int* __restrict__ src,
                               const int* __restrict__ dst,
                               const float* __restrict__ dinv,
                               float* __restrict__ agg, int E) {
  const int lane   = threadIdx.x & 31;
  const int wave   = (blockIdx.x * blockDim.x + threadIdx.x) >> 5;
  const int nwaves = (gridDim.x * blockDim.x) >> 5;
  for (int e = wave; e < E; e += nwaves) {
    const int   s   = src[e];
    const int   d   = dst[e];
    const float nrm = dinv[s] * dinv[d];
    #pragma unroll
    for (int f0 = 0; f0 < F; f0 += 64) {
      const int f = f0 + lane * 2;
      if (f < F) {
        const float2 v = *reinterpret_cast<const float2*>(h + (size_t)s * D_FEAT + f);
        atomicAdd(&agg[(size_t)d * D_FEAT + f],     v.x * nrm);
        atomicAdd(&agg[(size_t)d * D_FEAT + f + 1], v.y * nrm);
      }
    }
  }
}

// -------------------- self-loop + bias + ReLU (in place) -------------------
// agg[n,f] = relu(agg[n,f] + h[n,f]*dinv[n]^2 + b[f])
template <int F>
__global__ void finalize_kernel(float* __restrict__ agg, const float* __restrict__ h,
                                const float* __restrict__ dinv,
                                const float* __restrict__ b) {
  const int i = blockIdx.x * blockDim.x + threadIdx.x;
  const int n = i / F;                     // constant divisor
  const int f = i - n * F;
  if (n >= N_NODES) return;
  const float di = dinv[n];
  const float v  = agg[(size_t)n * D_FEAT + f] + h[(size_t)n * D_FEAT + f] * di * di + b[f];
  agg[(size_t)n * D_FEAT + f] = fmaxf(v, 0.0f);
}

// ------------------------------ log-softmax --------------------------------
__global__ void logsoftmax_kernel(const float* __restrict__ agg, float* __restrict__ out) {
  const int n = blockIdx.x * blockDim.x + threadIdx.x;
  if (n >= N_NODES) return;
  const float* r = agg + (size_t)n * D_FEAT;
  float mx = -3.0e38f;
  #pragma unroll
  for (int c = 0; c < N_CLASSES; ++c) mx = fmaxf(mx, r[c]);
  float s = 0.0f;
  #pragma unroll
  for (int c = 0; c < N_CLASSES; ++c) s += __expf(r[c] - mx);
  const float ls = __logf(s) + mx;
  #pragma unroll
  for (int c = 0; c < N_CLASSES; ++c) out[(size_t)n * N_CLASSES + c] = r[c] - ls;
}

// -------------------------------- launcher ---------------------------------

extern "C" void kernel_launch(void* const* d_in, const int* in_sizes, int n_in,
                              void* d_out, int out_size, void* d_ws, size_t ws_size,
                              hipStream_t stream) {
  const float* x  = (const float*)d_in[0];
  const int*   ei = (const int*)  d_in[1];   // [2, E] int32
  const float* W1 = (const float*)d_in[2];
  const float* b1 = (const float*)d_in[3];
  const float* W2 = (const float*)d_in[4];
  const float* b2 = (const float*)d_in[5];
  const float* W3 = (const float*)d_in[6];
  const float* b3 = (const float*)d_in[7];

  const int  E   = in_sizes[1] / 2;
  const int* src = ei;
  const int* dst = ei + E;

  // workspace layout (floats):
  //   dinv[N] | h[N*64] | agg[N*64] | Wt1[64*64] | Wt2[64*64] | Wt3[48*64]
  float* dinv = (float*)d_ws;
  float* h    = dinv + N_NODES;
  float* agg  = h    + (size_t)N_NODES * D_FEAT;
  float* Wt1  = agg  + (size_t)N_NODES * D_FEAT;
  float* Wt2  = Wt1  + 64 * 64;
  float* Wt3  = Wt2  + 64 * 64;
  float* out  = (float*)d_out;

  const int BT = 256;

  // degrees -> dinv (self-loop folded in as +1)
  zero_f32<<<(N_NODES + BT - 1) / BT, BT, 0, stream>>>(dinv, N_NODES);
  deg_kernel<<<(E + BT - 1) / BT, BT, 0, stream>>>(dst, dinv, E);
  dinv_kernel<<<(N_NODES + BT - 1) / BT, BT, 0, stream>>>(dinv, N_NODES);

  // transpose + pad weights: Wt[n][k], n padded to 16-tile multiple
  transpose_pad_w<<<(64 * 64 + BT - 1) / BT, BT, 0, stream>>>(W1, Wt1, 64, 64);
  transpose_pad_w<<<(64 * 64 + BT - 1) / BT, BT, 0, stream>>>(W2, Wt2, 64, 64);
  transpose_pad_w<<<(48 * 64 + BT - 1) / BT, BT, 0, stream>>>(W3, Wt3, 40, 48);

  const float* act    = x;
  const float* Wts[3] = {Wt1, Wt2, Wt3};
  const float* bs[3]  = {b1, b2, b3};
  const int    Fs[3]  = {D_FEAT, D_FEAT, N_CLASSES};

  for (int l = 0; l < 3; ++l) {
    const int F      = Fs[l];
    const int tilesN = (F + 15) / 16;      // 4, 4, 3
    const int waves  = TILES_M * tilesN;
    const int blocks = (waves * 32 + BT - 1) / BT;

    gemm_wmma<<<blocks, BT, 0, stream>>>(act, Wts[l], h, tilesN);
    zero_f32<<<((N_NODES * D_FEAT) + BT - 1) / BT, BT, 0, stream>>>(agg, N_NODES * D_FEAT);

    const int sblocks = (int)(((size_t)E * 32 + BT - 1) / BT);  // one wave/edge
    if (F == D_FEAT) {
      scatter_kernel<D_FEAT><<<sblocks, BT, 0, stream>>>(h, src, dst, dinv, agg, E);
      finalize_kernel<D_FEAT><<<((N_NODES * D_FEAT) + BT - 1) / BT, BT, 0, stream>>>(agg, h, dinv, bs[l]);
    } else {
      scatter_kernel<N_CLASSES><<<sblocks, BT, 0, stream>>>(h, src, dst, dinv, agg, E);
      finalize_kernel<N_CLASSES><<<((N_NODES * N_CLASSES) + BT - 1) / BT, BT, 0, stream>>>(agg, h, dinv, bs[l]);
    }
    act = agg;   // in-place activation becomes next layer's input
  }

  logsoftmax_kernel<<<(N_NODES + BT - 1) / BT, BT, 0, stream>>>(agg, out);
}